// FourierKNOConv2d_283467841872
// MI455X (gfx1250) — compile-verified
//
#include <hip/hip_runtime.h>
#include <hip/hip_bf16.h>
#include <math.h>

// ---------------------------------------------------------------------------
// FourierKNOConv2d on MI455X (gfx1250): decimated DFTs as WMMA f32 GEMMs.
// Every GEMM wave computes a 32x32 output tile via 4 accumulators (2 A-frags x
// 2 B-frags -> 4 v_wmma_f32_16x16x4_f32 per K-step), halving loads per WMMA.
// Channel mix stages per-lane data into LDS with global_load_async_to_lds_b32.
// ---------------------------------------------------------------------------

typedef __attribute__((ext_vector_type(2))) float v2f;
typedef __attribute__((ext_vector_type(8))) float v8f;

#define PI2 6.283185307179586f

__device__ __forceinline__ v8f wmma4(v2f a, v2f b, v8f c) {
  // (neg_a, A, neg_b, B, c_mod, C, reuse_a, reuse_b)
  return __builtin_amdgcn_wmma_f32_16x16x4_f32(false, a, false, b, (short)0, c,
                                               false, false);
}

// ---------------------------------------------------------------------------
// DFT tables with ortho norms folded in.
//   Fw [256x64]: cols 0..31 = cos/256, 32..63 = -sin/256       (fwd W-axis)
//   A2 [128x512]: [[C,S],[-S,C]], 64 k1 modes re/im            (fwd H-axis)
//   A4 [512x128]: [[C,-S],[S,C]]                               (inv H-axis)
//   G  [64x256]: rows 0..31 wgt*cos/256, 32..63 = -wgt*sin/256 (inv W-axis)
// k1 mode list: m<32 -> m (positive freqs), m>=32 -> m+192 (rows 224..255).
// ---------------------------------------------------------------------------
__global__ void build_tables(float* __restrict__ Fw, float* __restrict__ A2,
                             float* __restrict__ A4, float* __restrict__ G) {
  int id = blockIdx.x * blockDim.x + threadIdx.x;  // 0..65535
  if (id < 256 * 64) {
    int w = id >> 6, j = id & 63;
    int k2 = j & 31;
    float th = PI2 * (float)((k2 * w) & 255) * (1.f / 256.f);
    Fw[id] = ((j < 32) ? cosf(th) : -sinf(th)) * (1.f / 256.f);
  }
  {  // A2: exactly 65536 elements
    int r = id >> 9, k = id & 511;
    int m = r & 63;
    int k1 = (m < 32) ? m : (m + 192);
    int h = k & 255;
    float th = PI2 * (float)((k1 * h) & 255) * (1.f / 256.f);
    float c = cosf(th), s = sinf(th);
    A2[id] = (r < 64) ? ((k < 256) ? c : s) : ((k < 256) ? -s : c);
  }
  {  // A4: exactly 65536 elements
    int hp = id >> 7, j = id & 127;
    int m = j & 63;
    int k1 = (m < 32) ? m : (m + 192);
    int h = hp & 255;
    float th = PI2 * (float)((k1 * h) & 255) * (1.f / 256.f);
    float c = cosf(th), s = sinf(th);
    A4[id] = (hp < 256) ? ((j < 64) ? c : -s) : ((j < 64) ? s : c);
  }
  if (id < 64 * 256) {
    int j = id >> 8, w = id & 255;
    int k2 = j & 31;
    float th = PI2 * (float)((k2 * w) & 255) * (1.f / 256.f);
    float wgt = ((k2 == 0) ? 1.f : 2.f) * (1.f / 256.f);
    G[id] = (j < 32) ? (wgt * cosf(th)) : (-wgt * sinf(th));
  }
}

// ---------------------------------------------------------------------------
// Circulant channel-mix table: fft(ortho) o diag(k^r) o ifft(ortho) over 32
// channels == circular conv with w[d] = (1/32) sum_f k^r[f] e^{+i2pi f d/32}.
// wtab layout: [s][m1][m2][d] x {re,im}
// ---------------------------------------------------------------------------
__global__ void build_wtab(const float* __restrict__ kern,
                           const float* __restrict__ rp,
                           float* __restrict__ wtab) {
  int id = blockIdx.x * blockDim.x + threadIdx.x;  // 65536 = 2*32*32*32
  int d = id & 31;
  int m2 = (id >> 5) & 31;
  int m1 = (id >> 10) & 31;
  int s = id >> 15;
  float r = rp[0];
  float sr = 0.f, si = 0.f;
  for (int f = 0; f < 32; ++f) {
    int base = (((s * 32 + f) * 32 + m1) * 32 + m2) * 2;
    float kr = kern[base], ki = kern[base + 1];
    float mag2 = kr * kr + ki * ki;
    float pr = 0.f, pi = 0.f;
    if (mag2 > 0.f) {  // complex power k^r = |k|^r * e^{i r arg(k)}
      float p = powf(mag2, 0.5f * r);
      float ang = r * atan2f(ki, kr);
      pr = p * cosf(ang);
      pi = p * sinf(ang);
    }
    float th = PI2 * (float)((f * d) & 31) * (1.f / 32.f);
    float ce = cosf(th), se = sinf(th);
    sr += pr * ce - pi * se;
    si += pr * se + pi * ce;
  }
  wtab[id * 2] = sr * (1.f / 32.f);
  wtab[id * 2 + 1] = si * (1.f / 32.f);
}

// ---------------------------------------------------------------------------
// K1: u = x(256x256) * Fw(256x64) per image. Block = 2 waves (64 thr) covers
// 32 rows x 64 cols; each wave does a 32x32 tile (4 accumulators).
// Output U[512][32]: rows 0..255 = Re(u), rows 256..511 = Im(u).
// ---------------------------------------------------------------------------
__global__ void __launch_bounds__(64) k1_fwd_w(const float* __restrict__ x,
                                               const float* __restrict__ Fw,
                                               float* __restrict__ U) {
  const int img = blockIdx.x, rg = blockIdx.y;  // rg < 8
  const int wave = threadIdx.x >> 5, lane = threadIdx.x & 31;
  const int half = lane >> 4, lm = lane & 15;
  const int r0 = rg * 32;
  const int c0 = wave * 32;
  const float* Ar0 = x + (size_t)img * 65536 + (size_t)(r0 + lm) * 256;
  const float* Ar1 = Ar0 + 16 * 256;
  v8f acc00 = {}, acc01 = {}, acc10 = {}, acc11 = {};
  for (int k = 0; k < 256; k += 4) {
    __builtin_prefetch(Ar0 + k + 128, 0, 1);
    __builtin_prefetch(Ar1 + k + 128, 0, 1);
    v2f a0 = *(const v2f*)(Ar0 + k + 2 * half);
    v2f a1 = *(const v2f*)(Ar1 + k + 2 * half);
    v2f b0, b1;
    const float* Brow0 = Fw + (k + 2 * half) * 64;
    b0.x = Brow0[c0 + lm];
    b0.y = Brow0[64 + c0 + lm];
    b1.x = Brow0[c0 + 16 + lm];
    b1.y = Brow0[64 + c0 + 16 + lm];
    acc00 = wmma4(a0, b0, acc00);
    acc01 = wmma4(a0, b1, acc01);
    acc10 = wmma4(a1, b0, acc10);
    acc11 = wmma4(a1, b1, acc11);
  }
  float* Ui = U + (size_t)img * (512 * 32);
  const int j0 = c0 + lm, j1 = c0 + 16 + lm;
  const int jc0 = j0 & 31, of0 = (j0 < 32) ? 0 : 256;
  const int jc1 = j1 & 31, of1 = (j1 < 32) ? 0 : 256;
#pragma unroll
  for (int q = 0; q < 8; ++q) {
    int h0 = r0 + q + 8 * half, h1 = h0 + 16;
    Ui[(of0 + h0) * 32 + jc0] = acc00[q];
    Ui[(of1 + h0) * 32 + jc1] = acc01[q];
    Ui[(of0 + h1) * 32 + jc0] = acc10[q];
    Ui[(of1 + h1) * 32 + jc1] = acc11[q];
  }
}

// ---------------------------------------------------------------------------
// K2: X[128x32] = A2(128x512) * U(512x32) per image. Block = 1 wave, 32x32.
// Rows 0..63 = Re, 64..127 = Im.
// ---------------------------------------------------------------------------
__global__ void __launch_bounds__(32) k2_fwd_h(const float* __restrict__ A2,
                                               const float* __restrict__ U,
                                               float* __restrict__ X) {
  const int img = blockIdx.x, rg = blockIdx.y;  // rg < 4
  const int lane = threadIdx.x & 31;
  const int half = lane >> 4, lm = lane & 15;
  const int r0 = rg * 32;
  const float* B = U + (size_t)img * (512 * 32);
  v8f acc00 = {}, acc01 = {}, acc10 = {}, acc11 = {};
  for (int k = 0; k < 512; k += 4) {
    v2f a0 = *(const v2f*)(A2 + (r0 + lm) * 512 + k + 2 * half);
    v2f a1 = *(const v2f*)(A2 + (r0 + 16 + lm) * 512 + k + 2 * half);
    const float* Brow = B + (k + 2 * half) * 32;
    v2f b0, b1;
    b0.x = Brow[lm];
    b0.y = Brow[32 + lm];
    b1.x = Brow[16 + lm];
    b1.y = Brow[32 + 16 + lm];
    acc00 = wmma4(a0, b0, acc00);
    acc01 = wmma4(a0, b1, acc01);
    acc10 = wmma4(a1, b0, acc10);
    acc11 = wmma4(a1, b1, acc11);
  }
  float* Xi = X + (size_t)img * (128 * 32);
#pragma unroll
  for (int q = 0; q < 8; ++q) {
    int ra = r0 + q + 8 * half, rb = ra + 16;
    Xi[ra * 32 + lm] = acc00[q];
    Xi[ra * 32 + 16 + lm] = acc01[q];
    Xi[rb * 32 + lm] = acc10[q];
    Xi[rb * 32 + 16 + lm] = acc11[q];
  }
}

// ---------------------------------------------------------------------------
// K3: per-frequency circulant channel mix: out[co] = sum_ci w[(co-ci)&31]*in[ci]
// One wave per (batch, mode m, m2). Lane ci DMAs its channel's (re,im) into
// LDS via CDNA5 async-to-LDS, then all lanes (lane==co) reduce from LDS.
// ---------------------------------------------------------------------------
__global__ void __launch_bounds__(32) k3_mix(const float* __restrict__ X,
                                             const float* __restrict__ wtab,
                                             float* __restrict__ Xm) {
  __shared__ float sh[64];  // [ci] x {re, im}
  const int id = blockIdx.x;     // 8 * 64 * 32
  const int m2 = id & 31;
  const int m = (id >> 5) & 63;  // 0..31 pos rows, 32..63 neg rows
  const int b = id >> 11;
  const int s = m >> 5, m1 = m & 31;
  const float* wt = wtab + (((s * 32 + m1) * 32 + m2) * 32) * 2;
  const int lane = threadIdx.x;  // = ci for staging, = co for compute

  // Stage this lane's channel (re, im) straight into LDS (ASYNCcnt path).
  const float* Xin = X + (size_t)(b * 32 + lane) * (128 * 32);
  unsigned long long gre = (unsigned long long)(size_t)(Xin + m * 32 + m2);
  unsigned long long gim = (unsigned long long)(size_t)(Xin + (64 + m) * 32 + m2);
  unsigned lre = (unsigned)(size_t)&sh[2 * lane];
  unsigned lim = (unsigned)(size_t)&sh[2 * lane + 1];
  asm volatile("global_load_async_to_lds_b32 %0, %1, off"
               :
               : "v"(lre), "v"(gre)
               : "memory");
  asm volatile("global_load_async_to_lds_b32 %0, %1, off"
               :
               : "v"(lim), "v"(gim)
               : "memory");
  asm volatile("s_wait_asynccnt 0" ::: "memory");  // whole wave's DMAs done

  float sr = 0.f, si = 0.f;
  for (int ci = 0; ci < 32; ++ci) {
    float xr = sh[2 * ci];
    float xi = sh[2 * ci + 1];
    int d = (lane - ci) & 31;
    float wr = wt[2 * d], wi = wt[2 * d + 1];
    sr += wr * xr - wi * xi;
    si += wr * xi + wi * xr;
  }
  float* Xo = Xm + (size_t)(b * 32 + lane) * (128 * 32);
  Xo[m * 32 + m2] = sr;
  Xo[(64 + m) * 32 + m2] = si;
}

// ---------------------------------------------------------------------------
// K4: V[512x32] = A4(512x128) * Xm(128x32) per image. Block = 1 wave, 32x32.
// V rows 0..255 = Re(v), 256..511 = Im(v).
// ---------------------------------------------------------------------------
__global__ void __launch_bounds__(32) k4_inv_h(const float* __restrict__ A4,
                                               const float* __restrict__ Xm,
                                               float* __restrict__ V) {
  const int img = blockIdx.x, rg = blockIdx.y;  // rg < 16
  const int lane = threadIdx.x & 31;
  const int half = lane >> 4, lm = lane & 15;
  const int r0 = rg * 32;
  const float* B = Xm + (size_t)img * (128 * 32);
  v8f acc00 = {}, acc01 = {}, acc10 = {}, acc11 = {};
  for (int k = 0; k < 128; k += 4) {
    v2f a0 = *(const v2f*)(A4 + (r0 + lm) * 128 + k + 2 * half);
    v2f a1 = *(const v2f*)(A4 + (r0 + 16 + lm) * 128 + k + 2 * half);
    const float* Brow = B + (k + 2 * half) * 32;
    v2f b0, b1;
    b0.x = Brow[lm];
    b0.y = Brow[32 + lm];
    b1.x = Brow[16 + lm];
    b1.y = Brow[32 + 16 + lm];
    acc00 = wmma4(a0, b0, acc00);
    acc01 = wmma4(a0, b1, acc01);
    acc10 = wmma4(a1, b0, acc10);
    acc11 = wmma4(a1, b1, acc11);
  }
  float* Vi = V + (size_t)img * (512 * 32);
#pragma unroll
  for (int q = 0; q < 8; ++q) {
    int ra = r0 + q + 8 * half, rb = ra + 16;
    Vi[ra * 32 + lm] = acc00[q];
    Vi[ra * 32 + 16 + lm] = acc01[q];
    Vi[rb * 32 + lm] = acc10[q];
    Vi[rb * 32 + 16 + lm] = acc11[q];
  }
}

// ---------------------------------------------------------------------------
// K5: y(256x256) = Vh(256x64) * G(64x256) per image, Vh[h][j] = (j<32?Re:Im).
// Block = 2 waves covers 32 rows x 64 cols; each wave a 32x32 tile.
// ---------------------------------------------------------------------------
__global__ void __launch_bounds__(64) k5_inv_w(const float* __restrict__ V,
                                               const float* __restrict__ G,
                                               float* __restrict__ y) {
  const int img = blockIdx.x, rg = blockIdx.y, cg = blockIdx.z;  // 8 x 4
  const int wave = threadIdx.x >> 5, lane = threadIdx.x & 31;
  const int half = lane >> 4, lm = lane & 15;
  const int r0 = rg * 32;
  const int c0 = cg * 64 + wave * 32;
  const float* Vi = V + (size_t)img * (512 * 32);
  v8f acc00 = {}, acc01 = {}, acc10 = {}, acc11 = {};
  for (int k = 0; k < 64; k += 4) {
    const int j0 = k + 2 * half;  // even; j0,j0+1 stay in same Re/Im block
    const int rofs = (j0 < 32) ? 0 : 256;
    v2f a0 = *(const v2f*)(Vi + (rofs + r0 + lm) * 32 + (j0 & 31));
    v2f a1 = *(const v2f*)(Vi + (rofs + r0 + 16 + lm) * 32 + (j0 & 31));
    const float* Brow = G + j0 * 256;
    v2f b0, b1;
    b0.x = Brow[c0 + lm];
    b0.y = Brow[256 + c0 + lm];
    b1.x = Brow[c0 + 16 + lm];
    b1.y = Brow[256 + c0 + 16 + lm];
    acc00 = wmma4(a0, b0, acc00);
    acc01 = wmma4(a0, b1, acc01);
    acc10 = wmma4(a1, b0, acc10);
    acc11 = wmma4(a1, b1, acc11);
  }
  float* yo = y + (size_t)img * 65536;
#pragma unroll
  for (int q = 0; q < 8; ++q) {
    int ra = r0 + q + 8 * half, rb = ra + 16;
    yo[ra * 256 + c0 + lm] = acc00[q];
    yo[ra * 256 + c0 + 16 + lm] = acc01[q];
    yo[rb * 256 + c0 + lm] = acc10[q];
    yo[rb * 256 + c0 + 16 + lm] = acc11[q];
  }
}

// ---------------------------------------------------------------------------
extern "C" void kernel_launch(void* const* d_in, const int* in_sizes, int n_in,
                              void* d_out, int out_size, void* d_ws,
                              size_t ws_size, hipStream_t stream) {
  const float* x = (const float*)d_in[0];     // [8,32,256,256]
  const float* kern = (const float*)d_in[1];  // [2,1,32,32,32,2]
  const float* r = (const float*)d_in[2];     // scalar
  float* out = (float*)d_out;                 // [8,32,256,256]

  float* ws = (float*)d_ws;  // ~43.1 MB of f32 scratch
  float* Fw = ws;            // 256*64        = 16384
  float* A2 = Fw + 16384;    // 128*512       = 65536
  float* A4 = A2 + 65536;    // 512*128       = 65536
  float* G = A4 + 65536;     // 64*256        = 16384
  float* wt = G + 16384;     // 2*32*32*32*2  = 131072
  float* U = wt + 131072;    // 256*512*32    = 4194304
  float* X = U + 4194304;    // 256*128*32    = 1048576
  float* Xm = X + 1048576;   // 256*128*32    = 1048576
  float* V = Xm + 1048576;   // 256*512*32    = 4194304

  hipLaunchKernelGGL(build_tables, dim3(256), dim3(256), 0, stream, Fw, A2, A4, G);
  hipLaunchKernelGGL(build_wtab, dim3(256), dim3(256), 0, stream, kern, r, wt);
  hipLaunchKernelGGL(k1_fwd_w, dim3(256, 8), dim3(64), 0, stream, x, Fw, U);
  hipLaunchKernelGGL(k2_fwd_h, dim3(256, 4), dim3(32), 0, stream, A2, U, X);
  hipLaunchKernelGGL(k3_mix, dim3(16384), dim3(32), 0, stream, X, wt, Xm);
  hipLaunchKernelGGL(k4_inv_h, dim3(256, 16), dim3(32), 0, stream, A4, Xm, V);
  hipLaunchKernelGGL(k5_inv_w, dim3(256, 8, 4), dim3(64), 0, stream, V, G, out);
}